// GraphConvNet_2147483648614
// MI455X (gfx1250) — compile-verified
//
#include <hip/hip_runtime.h>
#include <hip/hip_bf16.h>

#define SLOPE 0.01f
#define EPS_BN 1e-5f
#define DEMO 4

typedef float v2f __attribute__((ext_vector_type(2)));
typedef float v8f __attribute__((ext_vector_type(8)));

// ---------------- embedding gather: x0[i,f] = emb[ids[i], f], EMB=16 ----------------
__global__ void gather_embed(const float* __restrict__ emb, const int* __restrict__ ids,
                             float* __restrict__ x0, int n) {
    int i = blockIdx.x * blockDim.x + threadIdx.x;
    if (i >= n * 16) return;
    int node = i >> 4;
    int f = i & 15;
    x0[i] = emb[(size_t)ids[node] * 16 + f];
}

// ---------------- edge scatter: agg[dst] += x[src] * ew ----------------
template <int HIN>
__global__ void edge_scatter(const float* __restrict__ x, const int* __restrict__ src,
                             const int* __restrict__ dst, const float* __restrict__ ew,
                             float* __restrict__ agg, int ne) {
    const int CH = HIN / 4;
    int idx = blockIdx.x * blockDim.x + threadIdx.x;
    if (idx >= ne * CH) return;
    int e  = idx / CH;
    int c4 = (idx % CH) * 4;
    int s = src[e];
    int d = dst[e];
    float w = ew[e];
    const float4 v = *reinterpret_cast<const float4*>(x + (size_t)s * HIN + c4);
    float* o = agg + (size_t)d * HIN + c4;
    atomicAdd(o + 0, v.x * w);
    atomicAdd(o + 1, v.y * w);
    atomicAdd(o + 2, v.z * w);
    atomicAdd(o + 3, v.w * w);
}

// ---------------- fused conv GEMM: y = agg @ Wr^T + x @ Ws^T + b  (HOUT=128) --------
// One wave32 per 16x16 output tile using V_WMMA_F32_16X16X4_F32.
// A 16x4 f32 frag: lane L holds A[L%16][k0 + (L/16)*2 + {0,1}]
// B 4x16 f32 frag: lane L holds B[k0 + (L/16)*2 + {0,1}][L%16]  (B = W^T)
// C/D 16x16 f32:  VGPR r, lane L -> D[r + 8*(L/16)][L%16]
template <int HIN>
__global__ __launch_bounds__(256) void gemm_conv(const float* __restrict__ x,
                                                 const float* __restrict__ agg,
                                                 const float* __restrict__ Wr,
                                                 const float* __restrict__ Ws,
                                                 const float* __restrict__ bias,
                                                 float* __restrict__ y, int nrows) {
    const int wave = threadIdx.x >> 5;   // 0..7 -> output tile
    const int lane = threadIdx.x & 31;
    const int half = lane >> 4;          // 0/1
    const int l16  = lane & 15;
    const int row0 = blockIdx.x * 16;    // node tile base (nrows % 16 == 0)
    const int n0   = wave * 16;          // output feature base

    const float* ar  = agg + (size_t)(row0 + l16) * HIN;
    const float* xr  = x   + (size_t)(row0 + l16) * HIN;
    const float* wrr = Wr  + (size_t)(n0 + l16) * HIN;
    const float* wsr = Ws  + (size_t)(n0 + l16) * HIN;

    v8f c = {0.f, 0.f, 0.f, 0.f, 0.f, 0.f, 0.f, 0.f};
    #pragma unroll 4
    for (int k = 0; k < HIN; k += 4) {
        const int ko = k + half * 2;
        v2f a0 = {ar[ko], ar[ko + 1]};
        v2f b0 = {wrr[ko], wrr[ko + 1]};
        c = __builtin_amdgcn_wmma_f32_16x16x4_f32(false, a0, false, b0, (short)0, c, false, false);
        v2f a1 = {xr[ko], xr[ko + 1]};
        v2f b1 = {wsr[ko], wsr[ko + 1]};
        c = __builtin_amdgcn_wmma_f32_16x16x4_f32(false, a1, false, b1, (short)0, c, false, false);
    }

    const float bb = bias[n0 + l16];
    #pragma unroll
    for (int r = 0; r < 8; ++r) {
        const int mr = r + half * 8;
        y[(size_t)(row0 + mr) * 128 + n0 + l16] = c[r] + bb;
    }
}

// ---------------- BN over N rows, 128 features: partial sums ----------------
__global__ void bn_stats(const float* __restrict__ x, float* __restrict__ sum,
                         float* __restrict__ sumsq, int nrows) {
    const int f = threadIdx.x;  // blockDim.x == 128
    int rows_per_block = (nrows + gridDim.x - 1) / gridDim.x;
    int r0 = blockIdx.x * rows_per_block;
    int r1 = r0 + rows_per_block;
    if (r1 > nrows) r1 = nrows;
    float s = 0.f, s2 = 0.f;
    for (int r = r0; r < r1; ++r) {
        float v = x[(size_t)r * 128 + f];
        s += v;
        s2 += v * v;
    }
    atomicAdd(&sum[f], s);
    atomicAdd(&sumsq[f], s2);
}

__global__ void bn_apply_leaky(float* __restrict__ x, const float* __restrict__ sum,
                               const float* __restrict__ sumsq, const float* __restrict__ g,
                               const float* __restrict__ be, int nrows) {
    int i = blockIdx.x * blockDim.x + threadIdx.x;
    if (i >= nrows * 128) return;
    int f = i & 127;
    float inv_n = 1.0f / (float)nrows;
    float mu  = sum[f] * inv_n;
    float var = sumsq[f] * inv_n - mu * mu;   // biased variance (jnp.var)
    float v = g[f] * (x[i] - mu) * rsqrtf(var + EPS_BN) + be[f];
    x[i] = v > 0.f ? v : SLOPE * v;
}

// ---------------- graph pooling ----------------
__global__ void pool_count(const int* __restrict__ batch, float* __restrict__ cnt, int n) {
    int i = blockIdx.x * blockDim.x + threadIdx.x;
    if (i < n) atomicAdd(&cnt[batch[i]], 1.0f);
}

__global__ void pool_sum(const float* __restrict__ x, const int* __restrict__ batch,
                         float* __restrict__ gfsum, int n) {
    int i = blockIdx.x * blockDim.x + threadIdx.x;
    if (i >= n * 128) return;
    int node = i >> 7;
    int f = i & 127;
    atomicAdd(&gfsum[(size_t)batch[node] * 128 + f], x[i]);
}

// h[:, :128] = gfsum / max(cnt,1);  h[:, 128:256] = demographics @ dW^T + db
__global__ void build_h(const float* __restrict__ gfsum, const float* __restrict__ cnt,
                        const float* __restrict__ demog, const float* __restrict__ dW,
                        const float* __restrict__ db, float* __restrict__ h, int ng) {
    int i = blockIdx.x * blockDim.x + threadIdx.x;
    if (i >= ng * 256) return;
    int grp = i >> 8;
    int f = i & 255;
    if (f < 128) {
        h[i] = gfsum[(size_t)grp * 128 + f] / fmaxf(cnt[grp], 1.0f);
    } else {
        int o = f - 128;
        float a = db[o];
        #pragma unroll
        for (int k = 0; k < DEMO; ++k) a += demog[grp * DEMO + k] * dW[o * DEMO + k];
        h[i] = a;
    }
}

// ---------------- tiny head GEMMs / BN ----------------
__global__ void head_gemm(const float* __restrict__ in, const float* __restrict__ W,
                          const float* __restrict__ b, float* __restrict__ out,
                          int M, int Kd, int Nd) {
    int i = blockIdx.x * blockDim.x + threadIdx.x;
    if (i >= M * Nd) return;
    int row = i / Nd;
    int o = i % Nd;
    float a = b[o];
    for (int k = 0; k < Kd; ++k) a += in[row * Kd + k] * W[o * Kd + k];
    out[i] = a;
}

__global__ void head_bn_leaky(float* __restrict__ x, const float* __restrict__ g,
                              const float* __restrict__ be, int rows, int F) {
    int f = threadIdx.x;
    if (f >= F) return;
    float s = 0.f, s2 = 0.f;
    for (int r = 0; r < rows; ++r) {
        float v = x[r * F + f];
        s += v;
        s2 += v * v;
    }
    float inv_n = 1.0f / (float)rows;
    float mu = s * inv_n;
    float var = s2 * inv_n - mu * mu;
    float inv = rsqrtf(var + EPS_BN);
    for (int r = 0; r < rows; ++r) {
        float v = g[f] * (x[r * F + f] - mu) * inv + be[f];
        x[r * F + f] = v > 0.f ? v : SLOPE * v;
    }
}

extern "C" void kernel_launch(void* const* d_in, const int* in_sizes, int n_in,
                              void* d_out, int out_size, void* d_ws, size_t ws_size,
                              hipStream_t stream) {
    // Inputs in setup_inputs() order
    const int*   x_ids = (const int*)d_in[0];
    const int*   eidx  = (const int*)d_in[1];   // (2, E)
    const float* eattr = (const float*)d_in[2]; // (E, 1)
    const int*   batch = (const int*)d_in[3];
    const float* demog = (const float*)d_in[4]; // (G, 4)
    const float* emb   = (const float*)d_in[5];
    const float* Wr0 = (const float*)d_in[6];
    const float* Ws0 = (const float*)d_in[7];
    const float* b0  = (const float*)d_in[8];
    const float* Wr1 = (const float*)d_in[9];
    const float* Ws1 = (const float*)d_in[10];
    const float* b1  = (const float*)d_in[11];
    const float* Wr2 = (const float*)d_in[12];
    const float* Ws2 = (const float*)d_in[13];
    const float* b2  = (const float*)d_in[14];
    const float* g0  = (const float*)d_in[15];
    const float* be0 = (const float*)d_in[16];
    const float* g1  = (const float*)d_in[17];
    const float* be1 = (const float*)d_in[18];
    const float* dW  = (const float*)d_in[19];
    const float* db  = (const float*)d_in[20];
    const float* cW0 = (const float*)d_in[21];
    const float* cb0 = (const float*)d_in[22];
    const float* cg0 = (const float*)d_in[23];
    const float* cbe0= (const float*)d_in[24];
    const float* cW1 = (const float*)d_in[25];
    const float* cb1 = (const float*)d_in[26];
    const float* cg1 = (const float*)d_in[27];
    const float* cbe1= (const float*)d_in[28];
    const float* cW2 = (const float*)d_in[29];
    const float* cb2 = (const float*)d_in[30];

    const int nN = in_sizes[0];          // 50000 (multiple of 16)
    const int nE = in_sizes[1] / 2;      // 800000
    const int nG = in_sizes[4] / DEMO;   // 128

    // workspace layout (floats)
    float* ws    = (float*)d_ws;
    float* bufA  = ws;                                // N*128
    float* bufB  = bufA + (size_t)nN * 128;           // N*128
    float* bufC  = bufB + (size_t)nN * 128;           // N*128
    float* sum   = bufC + (size_t)nN * 128;           // 128
    float* sumsq = sum + 128;                         // 128
    float* cnt   = sumsq + 128;                       // G
    float* gfsum = cnt + nG;                          // G*128
    float* h     = gfsum + (size_t)nG * 128;          // G*256
    float* h0    = h + (size_t)nG * 256;              // G*64
    float* h1    = h0 + (size_t)nG * 64;              // G*32

    const int* src = eidx;
    const int* dst = eidx + nE;

    // ---- layer 0 (EMB=16 -> 128) ----
    gather_embed<<<(nN * 16 + 255) / 256, 256, 0, stream>>>(emb, x_ids, bufA, nN);
    hipMemsetAsync(bufB, 0, (size_t)nN * 16 * sizeof(float), stream);
    edge_scatter<16><<<(nE * 4 + 255) / 256, 256, 0, stream>>>(bufA, src, dst, eattr, bufB, nE);
    gemm_conv<16><<<nN / 16, 256, 0, stream>>>(bufA, bufB, Wr0, Ws0, b0, bufC, nN);
    hipMemsetAsync(sum, 0, 2 * 128 * sizeof(float), stream);
    bn_stats<<<256, 128, 0, stream>>>(bufC, sum, sumsq, nN);
    bn_apply_leaky<<<(nN * 128 + 255) / 256, 256, 0, stream>>>(bufC, sum, sumsq, g0, be0, nN);

    // ---- layer 1 (128 -> 128) ----
    hipMemsetAsync(bufA, 0, (size_t)nN * 128 * sizeof(float), stream);
    edge_scatter<128><<<(nE * 32 + 255) / 256, 256, 0, stream>>>(bufC, src, dst, eattr, bufA, nE);
    gemm_conv<128><<<nN / 16, 256, 0, stream>>>(bufC, bufA, Wr1, Ws1, b1, bufB, nN);
    hipMemsetAsync(sum, 0, 2 * 128 * sizeof(float), stream);
    bn_stats<<<256, 128, 0, stream>>>(bufB, sum, sumsq, nN);
    bn_apply_leaky<<<(nN * 128 + 255) / 256, 256, 0, stream>>>(bufB, sum, sumsq, g1, be1, nN);

    // ---- layer 2 (128 -> 128, no BN/act) ----
    hipMemsetAsync(bufC, 0, (size_t)nN * 128 * sizeof(float), stream);
    edge_scatter<128><<<(nE * 32 + 255) / 256, 256, 0, stream>>>(bufB, src, dst, eattr, bufC, nE);
    gemm_conv<128><<<nN / 16, 256, 0, stream>>>(bufB, bufC, Wr2, Ws2, b2, bufA, nN);

    // ---- pooling + demographics ----
    hipMemsetAsync(cnt, 0, (size_t)(nG + (size_t)nG * 128) * sizeof(float), stream);
    pool_count<<<(nN + 255) / 256, 256, 0, stream>>>(batch, cnt, nN);
    pool_sum<<<(nN * 128 + 255) / 256, 256, 0, stream>>>(bufA, batch, gfsum, nN);
    build_h<<<(nG * 256 + 255) / 256, 256, 0, stream>>>(gfsum, cnt, demog, dW, db, h, nG);

    // ---- classifier head ----
    head_gemm<<<(nG * 64 + 255) / 256, 256, 0, stream>>>(h, cW0, cb0, h0, nG, 256, 64);
    head_bn_leaky<<<1, 64, 0, stream>>>(h0, cg0, cbe0, nG, 64);
    head_gemm<<<(nG * 32 + 255) / 256, 256, 0, stream>>>(h0, cW1, cb1, h1, nG, 64, 32);
    head_bn_leaky<<<1, 32, 0, stream>>>(h1, cg1, cbe1, nG, 32);
    head_gemm<<<1, 256, 0, stream>>>(h1, cW2, cb2, (float*)d_out, nG, 32, 2);
}